// RNN_20572893348005
// MI455X (gfx1250) — compile-verified
//
#include <hip/hip_runtime.h>
#include <hip/hip_bf16.h>

#define BB 64
#define TT 1024
#define DD 128
#define HH 256

typedef float v2f __attribute__((ext_vector_type(2)));
typedef float v4f __attribute__((ext_vector_type(4)));
typedef float v8f __attribute__((ext_vector_type(8)));

__device__ __forceinline__ v8f wmma_f32_16x16x4(v2f a, v2f b, v8f c) {
    // D = A(16x4,f32) * B(4x16,f32) + C(16x16,f32)
    return __builtin_amdgcn_wmma_f32_16x16x4_f32(false, a, false, b, (short)0, c,
                                                 false, false);
}

// Hardware tanh (CDNA5 V_TANH_F32). Builtin path lets the compiler handle the
// TRANS-op hazard; the asm fallback bakes in the required independent-op gap.
__device__ __forceinline__ float hw_tanh(float x) {
#if __has_builtin(__builtin_amdgcn_tanhf)
    return __builtin_amdgcn_tanhf(x);
#else
    float r;
    asm volatile("v_tanh_f32 %0, %1\n\tv_nop\n\tv_nop" : "=v"(r) : "v"(x));
    return r;
#endif
}

// -------------------------------------------------------------------------
// Input projection: out[m, n] = sum_k in[m,k] * W[n,k] + bias[n]
// M = B*T rows, N = H = 256, K = 128 or 256. One WG = 16 M-rows, 8 waves,
// each wave computes two 16x16 N-tiles with f32 WMMA.
// -------------------------------------------------------------------------
template <int K>
__global__ __launch_bounds__(256) void proj_kernel(const float* __restrict__ in,
                                                   const float* __restrict__ W,
                                                   const float* __restrict__ bias,
                                                   float* __restrict__ out) {
    constexpr int KP = K + 4;  // padded row stride (bank-conflict free)
    __shared__ float a_lds[16 * KP];
    const int tid = threadIdx.x;
    const int m0  = blockIdx.x * 16;

    // Stage the 16xK A tile into LDS, coalesced float4 loads.
    constexpr int NUM4 = 16 * K / 4;  // 512 (K=128) or 1024 (K=256)
    const v4f* in4 = (const v4f*)(in + (size_t)m0 * K);
#pragma unroll
    for (int j = 0; j < NUM4 / 256; ++j) {
        int idx = tid + j * 256;
        int m   = idx / (K / 4);
        int kq  = idx % (K / 4);
        *(v4f*)&a_lds[m * KP + kq * 4] = in4[idx];
    }
    __syncthreads();

    const int wave = tid >> 5;
    const int lane = tid & 31;
    const int ll   = lane & 15;
    const int lh   = lane >> 4;
    const int n0   = wave * 32;  // two N tiles: n0, n0+16

    v8f c0, c1;
    const float bi0 = bias[n0 + ll];
    const float bi1 = bias[n0 + 16 + ll];
#pragma unroll
    for (int g = 0; g < 8; ++g) { c0[g] = bi0; c1[g] = bi1; }

    const float* a_base  = &a_lds[ll * KP + 2 * lh];
    const float* b0_base = &W[(size_t)(n0 + ll) * K + 2 * lh];
    const float* b1_base = b0_base + (size_t)16 * K;

#pragma unroll 8
    for (int kk = 0; kk < K / 4; ++kk) {
        v2f a  = *(const v2f*)(a_base + kk * 4);
        v2f b0 = *(const v2f*)(b0_base + kk * 4);
        v2f b1 = *(const v2f*)(b1_base + kk * 4);
        c0 = wmma_f32_16x16x4(a, b0, c0);
        c1 = wmma_f32_16x16x4(a, b1, c1);
    }

#pragma unroll
    for (int g = 0; g < 8; ++g) {
        int m = g + 8 * lh;
        out[(size_t)(m0 + m) * HH + n0 + ll]      = c0[g];
        out[(size_t)(m0 + m) * HH + n0 + 16 + ll] = c1[g];
    }
}

// -------------------------------------------------------------------------
// Recurrence: h_t = tanh(xW[:,t,:] + h_{t-1} @ W_hh^T + b_hh)
// One WG per 16-batch tile (4 WGs, one per WGP). W_hh (256KB, padded) and the
// h tile (16x256) stay resident in LDS for all 1024 steps. 16 waves, each
// owning one 16-column output tile. xW loads for step t+1 are issued during
// step t (software pipeline) so HBM latency hides under the WMMA chain.
// Two accumulators split the K=256 reduction for XDL pipeline ILP.
// -------------------------------------------------------------------------
__global__ __launch_bounds__(512) void rec_kernel(const float* __restrict__ xW,
                                                  float* __restrict__ hout,
                                                  const float* __restrict__ Whh,
                                                  const float* __restrict__ bhh) {
    constexpr int WP = HH + 4;  // padded row: 260 floats
    extern __shared__ float smem[];
    float* w_lds = smem;             // [HH][WP]
    float* h_lds = smem + HH * WP;   // [16][WP]

    const int tid = threadIdx.x;

    // Stage W_hh (65536 floats) into padded LDS, coalesced float4.
    const v4f* w4 = (const v4f*)Whh;
#pragma unroll
    for (int j = 0; j < (HH * HH / 4) / 512; ++j) {  // 32 iters
        int idx = tid + j * 512;
        int n   = idx / (HH / 4);
        int kq  = idx % (HH / 4);
        *(v4f*)&w_lds[n * WP + kq * 4] = w4[idx];
    }
    for (int i = tid; i < 16 * WP; i += 512) h_lds[i] = 0.0f;  // h0 = 0

    const int wave = tid >> 5;
    const int lane = tid & 31;
    const int ll   = lane & 15;
    const int lh   = lane >> 4;
    const int n0   = wave * 16;
    const int b0   = blockIdx.x * 16;
    const float bias_n = bhh[n0 + ll];
    __syncthreads();

    const float* wbase = &w_lds[(n0 + ll) * WP + 2 * lh];
    const float* abase = &h_lds[ll * WP + 2 * lh];

    // Per-lane base into xW for (batch m, col n0+ll); row m adds m*T*H.
    const float* xbase = xW + (size_t)b0 * TT * HH + n0 + ll;

    // Preload xW for t = 0.
    float pre[8];
#pragma unroll
    for (int g = 0; g < 8; ++g)
        pre[g] = xbase[(size_t)(g + 8 * lh) * TT * HH];

    for (int t = 0; t < TT; ++t) {
        v8f c0, c1;
#pragma unroll
        for (int g = 0; g < 8; ++g) {
            c0[g] = pre[g] + bias_n;
            c1[g] = 0.0f;
        }

        // Issue next step's xW loads now; they complete under the WMMA chain.
        if (t + 1 < TT) {
#pragma unroll
            for (int g = 0; g < 8; ++g)
                pre[g] = xbase[(size_t)(g + 8 * lh) * TT * HH +
                               (size_t)(t + 1) * HH];
        }

        // h_{t-1} [16x256] @ W_hh^T tile [256x16], two accumulator chains.
#pragma unroll 16
        for (int kk = 0; kk < HH / 8; ++kk) {  // 32 iters, 2 K-steps each
            v2f a0 = *(const v2f*)(abase + kk * 8);
            v2f b0 = *(const v2f*)(wbase + kk * 8);
            v2f a1 = *(const v2f*)(abase + kk * 8 + 4);
            v2f b1 = *(const v2f*)(wbase + kk * 8 + 4);
            c0 = wmma_f32_16x16x4(a0, b0, c0);
            c1 = wmma_f32_16x16x4(a1, b1, c1);
        }

        __syncthreads();  // all reads of h_lds complete
#pragma unroll
        for (int g = 0; g < 8; ++g) {
            int   m = g + 8 * lh;
            float v = hw_tanh(c0[g] + c1[g]);
            h_lds[m * WP + n0 + ll] = v;
            hout[((size_t)(b0 + m) * TT + t) * HH + n0 + ll] = v;
        }
        __syncthreads();  // h_t visible to all waves
    }
}

// -------------------------------------------------------------------------
// [B, T, H] -> [B, H, T] transpose with LDS tiling (keeps both sides
// coalesced; direct scattered stores from rec would shatter cachelines).
// -------------------------------------------------------------------------
__global__ __launch_bounds__(256) void transpose_kernel(const float* __restrict__ in,
                                                        float* __restrict__ out) {
    __shared__ float tile[32][33];
    const int b  = blockIdx.z;
    const int t0 = blockIdx.x * 32;
    const int h0 = blockIdx.y * 32;
    const int tx = threadIdx.x, ty = threadIdx.y;
#pragma unroll
    for (int j = 0; j < 4; ++j) {
        int t = t0 + ty + j * 8;
        tile[ty + j * 8][tx] = in[((size_t)b * TT + t) * HH + h0 + tx];
    }
    __syncthreads();
#pragma unroll
    for (int j = 0; j < 4; ++j) {
        int h = h0 + ty + j * 8;
        out[((size_t)b * HH + h) * TT + t0 + tx] = tile[tx][ty + j * 8];
    }
}

extern "C" void kernel_launch(void* const* d_in, const int* in_sizes, int n_in,
                              void* d_out, int out_size, void* d_ws, size_t ws_size,
                              hipStream_t stream) {
    const float* x     = (const float*)d_in[0];
    const float* W_ih0 = (const float*)d_in[1];
    const float* W_hh0 = (const float*)d_in[2];
    const float* b_ih0 = (const float*)d_in[3];
    const float* b_hh0 = (const float*)d_in[4];
    const float* W_ih1 = (const float*)d_in[5];
    const float* W_hh1 = (const float*)d_in[6];
    const float* b_ih1 = (const float*)d_in[7];
    const float* b_hh1 = (const float*)d_in[8];

    float* buf0 = (float*)d_ws;                       // [B,T,H]
    float* buf1 = buf0 + (size_t)BB * TT * HH;        // [B,T,H]

    const size_t recLds = (size_t)(HH + 16) * (HH + 4) * sizeof(float);  // ~283 KB
    (void)hipFuncSetAttribute((const void*)rec_kernel,
                              hipFuncAttributeMaxDynamicSharedMemorySize,
                              (int)recLds);

    // Layer 0
    proj_kernel<DD><<<dim3(BB * TT / 16), 256, 0, stream>>>(x, W_ih0, b_ih0, buf0);
    rec_kernel<<<dim3(BB / 16), 512, recLds, stream>>>(buf0, buf1, W_hh0, b_hh0);
    // Layer 1
    proj_kernel<HH><<<dim3(BB * TT / 16), 256, 0, stream>>>(buf1, W_ih1, b_ih1, buf0);
    rec_kernel<<<dim3(BB / 16), 512, recLds, stream>>>(buf0, buf1, W_hh1, b_hh1);
    // [B,T,H] -> [B,H,T]
    transpose_kernel<<<dim3(TT / 32, HH / 32, BB), dim3(32, 8), 0, stream>>>(
        buf1, (float*)d_out);
}